// BlockDecomposition_3367254360146
// MI455X (gfx1250) — compile-verified
//
#include <hip/hip_runtime.h>

// ---------------------------------------------------------------------------
// RGCN block-diagonal message passing for MI455X (gfx1250, wave32).
//   out[n] = keep[n] * (xb[n] @ blocks[16])  +  sum_{edges e: tgt=n} ew[e] * (xb[src] @ blocks[rel[e]])
// Edges symmetrized (2E directed messages), binned by relation (bins padded to
// multiples of 16 with weight-0 dummies). Each wave takes 16 same-relation
// edges; the 16x128 message tile = 8 x (16x16) tiles, each built from
// 4 x v_wmma_f32_16x16x4_f32 (W is block-diagonal in K). Edge weight is folded
// into the A operand. Results scatter via global_atomic_add_f32 addressed as
// SGPR-base + 32-bit VGPR offset (out fits in 4 GB), so per-atomic address
// math is one v_mad_u32 per tile, with 16*nt*4 folded into the immediate.
// ---------------------------------------------------------------------------

typedef __attribute__((ext_vector_type(2))) float v2f;
typedef __attribute__((ext_vector_type(8))) float v8f;

#define IDIM 128
#define NBLK 32
#define BS 4
#define NRELW 17                        // 16 relations + self block
#define WTBL (NRELW * NBLK * BS * BS)   // 8704 floats = 34 KB

// ---- self loop: fully initializes out ------------------------------------
__global__ void self_kernel(const float* __restrict__ x,
                            const unsigned char* __restrict__ keep,
                            const float* __restrict__ blocks,
                            float* __restrict__ out, int n_nodes) {
    int n = blockIdx.x;
    int j = threadIdx.x;              // 0..127 output column
    if (n >= n_nodes) return;
    int b = j >> 2, jc = j & 3;
    float acc = 0.f;
    if (keep[n]) {
        const float* w = blocks + ((16 * NBLK + b) * 16);   // blocks[-1][b]
        const float* xr = x + (size_t)n * IDIM + b * 4;
        #pragma unroll
        for (int i = 0; i < 4; ++i) acc += xr[i] * w[i * 4 + jc];
    }
    out[(size_t)n * IDIM + j] = acc;
}

// ---- binning passes -------------------------------------------------------
__global__ void init_kernel(int* ssrc, int* stgt, float* sew, int* binCount, int pad2e) {
    for (int i = blockIdx.x * blockDim.x + threadIdx.x; i < pad2e;
         i += gridDim.x * blockDim.x) {
        ssrc[i] = 0; stgt[i] = 0; sew[i] = 0.f;
    }
    if (blockIdx.x == 0 && threadIdx.x < 16) binCount[threadIdx.x] = 0;
}

__global__ void count_kernel(const int* __restrict__ etype, int* binCount, int E) {
    for (int j = blockIdx.x * blockDim.x + threadIdx.x; j < E;
         j += gridDim.x * blockDim.x)
        atomicAdd(&binCount[etype[j]], 2);   // each edge contributes both directions
}

__global__ void scan_kernel(const int* __restrict__ binCount, int* binStart,
                            int* cursor, int* tileRel, int nTiles) {
    __shared__ int s_start[17];
    if (threadIdx.x == 0) {
        int run = 0;
        for (int r = 0; r < 16; ++r) {
            s_start[r] = run; binStart[r] = run; cursor[r] = run;
            run += ((binCount[r] + 15) >> 4) << 4;   // pad bin to multiple of 16
        }
        s_start[16] = run; binStart[16] = run;
    }
    __syncthreads();
    for (int t = threadIdx.x; t < nTiles; t += blockDim.x) {
        int e16 = t * 16, r = 0;
        while (r < 15 && e16 >= s_start[r + 1]) ++r;
        tileRel[t] = r;                 // tiles past the end map to rel 15 w/ zero weights
    }
}

__global__ void scatter_kernel(const int* __restrict__ source, const int* __restrict__ target,
                               const int* __restrict__ etype, const float* __restrict__ ew,
                               int* cursor, int* ssrc, int* stgt, float* sew, int E) {
    for (int j = blockIdx.x * blockDim.x + threadIdx.x; j < 2 * E;
         j += gridDim.x * blockDim.x) {
        int i = (j < E) ? j : j - E;
        int s = (j < E) ? source[i] : target[i];
        int t = (j < E) ? target[i] : source[i];
        int r = etype[i];
        int pos = atomicAdd(&cursor[r], 1);
        ssrc[pos] = s; stgt[pos] = t; sew[pos] = ew[i];
    }
}

// ---- WMMA edge kernel -----------------------------------------------------
// A layout (f32 16x4): lane l holds M=l&15; VGPR0 = K (l<16 ? 0 : 2), VGPR1 = K+1
// B layout (f32 4x16): lane l holds N=l&15; VGPR0 = K (l<16 ? 0 : 2), VGPR1 = K+1
// C layout:            VGPR v holds M = v + (l<16 ? 0 : 8), N = l&15
// LDS weight table is transposed to [r][b][j][k] so each lane's (k0,k0+1) pair
// is one ds_load_b64 (pairs merge into ds_load_2addr_b64). B's block-diagonal
// zero band is a {0,1}-float multiply, keeping EXEC untouched around the WMMAs.
__global__ void edge_wmma_kernel(const float* __restrict__ x,
                                 const float* __restrict__ blocks,
                                 const int* __restrict__ ssrc,
                                 const int* __restrict__ stgt,
                                 const float* __restrict__ sew,
                                 const int* __restrict__ tileRel,
                                 float* __restrict__ out, int nTiles) {
    __shared__ float wlds[WTBL];        // transposed: [(r*32+b)*16 + j*4 + k]
    for (int i = threadIdx.x; i < WTBL; i += blockDim.x) {
        int k = i & 3, j = (i >> 2) & 3, rb = i >> 4;
        wlds[(rb << 4) + j * 4 + k] = blocks[(rb << 4) + k * 4 + j];
    }
    __syncthreads();

    const int lane = threadIdx.x & 31;
    const int wave = threadIdx.x >> 5;
    const int wavesPerBlk = blockDim.x >> 5;
    const int n   = lane & 15;             // N column / M row owned by this lane
    const int k0  = (lane >> 4) << 1;      // K pair owned by this lane: 0 or 2
    const int mhi = (lane >> 4) << 3;      // C row offset: 0 or 8
    float maskf[4];
    #pragma unroll
    for (int kk = 0; kk < 4; ++kk) maskf[kk] = ((n >> 2) == kk) ? 1.f : 0.f;

    for (int tile = blockIdx.x * wavesPerBlk + wave; tile < nTiles;
         tile += gridDim.x * wavesPerBlk) {
        const int r = tileRel[tile];                 // uniform across wave
        const int e = tile * 16 + n;                 // this lane's edge (lanes 16-31 mirror)
        const int src = ssrc[e];
        const int tgt = stgt[e];
        const float w = sew[e];                      // folded into A below
        const float* xrow = x + (size_t)src * IDIM + k0;
        const float* wrel = wlds + ((r * NBLK) << 4) + ((n & 3) << 2) + k0;

        // 32-bit scatter offsets: one v_mad_u32 each, hoisted out of the nt loop
        unsigned offv[8];
        #pragma unroll
        for (int v = 0; v < 8; ++v) {
            unsigned tm = (unsigned)__shfl(tgt, v + mhi, 32);
            offv[v] = tm * (unsigned)IDIM + (unsigned)n;
        }

        #pragma unroll
        for (int nt = 0; nt < 8; ++nt) {
            v8f c = {};
            #pragma unroll
            for (int kk = 0; kk < 4; ++kk) {
                const int b = 4 * nt + kk;           // 4x4 block on the diagonal
                v2f av = *(const v2f*)(xrow + 16 * nt + 4 * kk);
                v2f wv = *(const v2f*)(wrel + (b << 4));
                v2f a, bv;
                a.x = av.x * w;  a.y = av.y * w;     // edge weight folded into A
                bv.x = wv.x * maskf[kk];             // zero outside 4-col band
                bv.y = wv.y * maskf[kk];
                c = __builtin_amdgcn_wmma_f32_16x16x4_f32(
                        false, a, false, bv, (short)0, c, false, false);
            }
            #pragma unroll
            for (int v = 0; v < 8; ++v)
                atomicAdd(out + offv[v] + 16u * nt, c[v]);   // saddr + voffset form
        }
    }
}

// ---- fallback (ws too small): plain VALU + atomics ------------------------
__global__ void edge_fallback_kernel(const float* __restrict__ x,
                                     const float* __restrict__ blocks,
                                     const int* __restrict__ source,
                                     const int* __restrict__ target,
                                     const int* __restrict__ etype,
                                     const float* __restrict__ ew,
                                     float* __restrict__ out, int E) {
    int total = 2 * E * NBLK;
    for (int tid = blockIdx.x * blockDim.x + threadIdx.x; tid < total;
         tid += gridDim.x * blockDim.x) {
        int j = tid >> 5, b = tid & 31;
        int i = (j < E) ? j : j - E;
        int s = (j < E) ? source[i] : target[i];
        int t = (j < E) ? target[i] : source[i];
        const float* w = blocks + ((etype[i] * NBLK + b) * 16);
        const float* xr = x + (size_t)s * IDIM + b * 4;
        float wgt = ew[i];
        #pragma unroll
        for (int jc = 0; jc < 4; ++jc) {
            float acc = 0.f;
            #pragma unroll
            for (int k = 0; k < 4; ++k) acc += xr[k] * w[k * 4 + jc];
            atomicAdd(&out[(size_t)t * IDIM + b * 4 + jc], acc * wgt);
        }
    }
}

// ---------------------------------------------------------------------------
extern "C" void kernel_launch(void* const* d_in, const int* in_sizes, int n_in,
                              void* d_out, int out_size, void* d_ws, size_t ws_size,
                              hipStream_t stream) {
    const float*         x      = (const float*)d_in[0];
    const unsigned char* keep   = (const unsigned char*)d_in[1];
    const int*           source = (const int*)d_in[2];
    const int*           target = (const int*)d_in[3];
    const int*           etype  = (const int*)d_in[4];
    const float*         ew     = (const float*)d_in[5];
    const float*         blocks = (const float*)d_in[6];
    float* out = (float*)d_out;

    const int N = in_sizes[0] / IDIM;
    const int E = in_sizes[2];

    // binned-edge workspace layout
    const int pad2e  = ((2 * E + 255) / 16) * 16;   // <= 240 padding slots needed
    const int nTiles = pad2e / 16;
    int*   ssrc    = (int*)d_ws;
    int*   stgt    = ssrc + pad2e;
    float* sew     = (float*)(stgt + pad2e);
    int*   tileRel = (int*)(sew + pad2e);
    int*   binCnt  = tileRel + nTiles;
    int*   binStart= binCnt + 16;
    int*   cursor  = binStart + 17;
    const size_t need = (size_t)(cursor + 16 - ssrc) * sizeof(int);

    self_kernel<<<N, IDIM, 0, stream>>>(x, keep, blocks, out, N);

    if (ws_size >= need) {
        init_kernel   <<<512, 256, 0, stream>>>(ssrc, stgt, sew, binCnt, pad2e);
        count_kernel  <<<256, 256, 0, stream>>>(etype, binCnt, E);
        scan_kernel   <<<1,   256, 0, stream>>>(binCnt, binStart, cursor, tileRel, nTiles);
        scatter_kernel<<<512, 256, 0, stream>>>(source, target, etype, ew,
                                                cursor, ssrc, stgt, sew, E);
        edge_wmma_kernel<<<256, 256, 0, stream>>>(x, blocks, ssrc, stgt, sew,
                                                  tileRel, out, nTiles);
    } else {
        int total = 2 * E * NBLK;
        edge_fallback_kernel<<<(total + 255) / 256, 256, 0, stream>>>(
            x, blocks, source, target, etype, ew, out, E);
    }
}